// Encoder_49959059587710
// MI455X (gfx1250) — compile-verified
//
#include <hip/hip_runtime.h>
#include <math.h>

// ---------------------------------------------------------------------------
// CDNA5 / gfx1250 implementation of the SplineConv GNN encoder.
// fp32 end-to-end; matrix work uses V_WMMA_F32_16X16X4_F32 (wave32).
// Each wave computes a 16x32 output tile (two WMMA accumulators sharing the
// A fragment) with branch-free, clamp-guarded loads in the K loop and a
// wave-uniform full/tail split in the store epilogue.
// ---------------------------------------------------------------------------

typedef float v2f __attribute__((ext_vector_type(2)));
typedef float v8f __attribute__((ext_vector_type(8)));

// Y[n, 9*co] = X[n, ci] @ [ W0 | W1 | ... | W7 | R ]   (column blocks of co)
// A layout (ISA 7.12.2, 32-bit A 16x4): lanes 0-15 hold M=lane, K={0,1};
// lanes 16-31 hold M=lane-16, K={2,3}. B mirrors with N in place of M.
// C/D: VGPR v holds M = v + (lane>=16 ? 8 : 0), N = lane&15.
__global__ void __launch_bounds__(128)
gemm_wmma_f32(const float* __restrict__ X, const float* __restrict__ W,
              const float* __restrict__ R, float* __restrict__ Y,
              int n, int ci, int co)
{
    const int ncols  = 9 * co;                      // always a multiple of 32
    const int lane   = threadIdx.x & 31;
    const int wave   = (blockIdx.x * blockDim.x + threadIdx.x) >> 5;
    const int tilesN = ncols >> 5;                  // 32-column supertiles
    const int tilesM = (n + 15) >> 4;
    const int tm = wave / tilesN;
    const int tn = wave - tm * tilesN;
    if (tm >= tilesM) return;                       // wave-uniform exit

    const int l15   = lane & 15;
    const int khalf = (lane >> 4) << 1;             // 0 or 2
    const int row   = tm * 16 + l15;
    const int rowc  = row < n ? row : (n - 1);      // clamp: loads stay valid
    const int col0  = tn * 32 + l15;                // two 16-col subtiles
    const int col1  = col0 + 16;

    // Column base pointers into [W0..W7 | R]; K-stride = co floats.
    const int    kb0 = col0 / co, cc0 = col0 - kb0 * co;
    const int    kb1 = col1 / co, cc1 = col1 - kb1 * co;
    const float* B0  = (kb0 < 8) ? (W + (size_t)kb0 * ci * co + cc0) : (R + cc0);
    const float* B1  = (kb1 < 8) ? (W + (size_t)kb1 * ci * co + cc1) : (R + cc1);
    const float* Arow = X + (size_t)rowc * ci;

    __builtin_prefetch(Arow, 0, 0);                 // global_prefetch_b8

    v8f acc0 = {0.f, 0.f, 0.f, 0.f, 0.f, 0.f, 0.f, 0.f};
    v8f acc1 = {0.f, 0.f, 0.f, 0.f, 0.f, 0.f, 0.f, 0.f};
    #pragma unroll 2
    for (int ka = khalf; ka < ci; ka += 4) {
        const v2f a = *(const v2f*)(Arow + ka);     // b64 load, unconditional
        v2f b0, b1;
        b0.x = B0[(size_t)ka * co];
        b0.y = B0[(size_t)(ka + 1) * co];
        b1.x = B1[(size_t)ka * co];
        b1.y = B1[(size_t)(ka + 1) * co];
        acc0 = __builtin_amdgcn_wmma_f32_16x16x4_f32(
            false, a, false, b0, (short)0, acc0, false, false);
        acc1 = __builtin_amdgcn_wmma_f32_16x16x4_f32(
            false, a, false, b1, (short)0, acc1, false, false);
    }

    // Epilogue: wave-uniform split; interior tiles take the unguarded path.
    const int mbase = tm * 16 + ((lane >> 4) ? 8 : 0);
    float* yr = Y + (size_t)mbase * ncols;          // marching row pointer
    if (tm * 16 + 16 <= n) {                        // full tile (common case)
        #pragma unroll
        for (int v = 0; v < 8; ++v) {
            yr[col0] = acc0[v];
            yr[col1] = acc1[v];
            yr += ncols;
        }
    } else {                                        // tail tile only
        #pragma unroll
        for (int v = 0; v < 8; ++v) {
            if (mbase + v < n) {
                yr[col0] = acc0[v];
                yr[col1] = acc1[v];
            }
            yr += ncols;
        }
    }
}

// Per (edge, 4-channel chunk): trilinear spline weights from edge_attr,
// gather from Ycat[src], reduce over the 8 kernels, atomic-add into ACC[dst].
__global__ void edge_accum(const float* __restrict__ Y,   // [n, 9*co]
                           const float* __restrict__ ea,  // [E, 3]
                           const int*   __restrict__ ei,  // [2, E]
                           float* __restrict__ ACC,       // [n, co]
                           float* __restrict__ CNT,       // [n]
                           int E, int co)
{
    const int  chunks = co >> 2;
    const long tid    = (long)blockIdx.x * blockDim.x + threadIdx.x;
    if (tid >= (long)E * chunks) return;
    const int e  = (int)(tid / chunks);
    const int c4 = (int)(tid - (long)e * chunks) << 2;

    const int s = ei[e];
    const int d = ei[E + e];
    const float px = ea[3 * e], py = ea[3 * e + 1], pz = ea[3 * e + 2];
    const float qx[2] = {1.f - px, px};
    const float qy[2] = {1.f - py, py};
    const float qz[2] = {1.f - pz, pz};

    const float* ys = Y + (size_t)s * 9 * co + c4;
    float s0 = 0.f, s1 = 0.f, s2 = 0.f, s3 = 0.f;
    #pragma unroll
    for (int k = 0; k < 8; ++k) {
        const float wk = qx[k & 1] * qy[(k >> 1) & 1] * qz[(k >> 2) & 1];
        const float4 yv = *(const float4*)(ys + (size_t)k * co);
        s0 = fmaf(wk, yv.x, s0);
        s1 = fmaf(wk, yv.y, s1);
        s2 = fmaf(wk, yv.z, s2);
        s3 = fmaf(wk, yv.w, s3);
    }
    float* a = ACC + (size_t)d * co + c4;
    atomicAdd(a + 0, s0);
    atomicAdd(a + 1, s1);
    atomicAdd(a + 2, s2);
    atomicAdd(a + 3, s3);
    if (c4 == 0) atomicAdd(&CNT[d], 1.f);
}

// h = elu(ACC/max(cnt,1) + Y_R + b); accumulate BN sum/sumsq. In-place on ACC.
__global__ void node_finalize(const float* __restrict__ Y,
                              float* __restrict__ ACC,      // in: sums, out: h
                              const float* __restrict__ CNT,
                              const float* __restrict__ b,
                              float* __restrict__ stats,    // [2*co]
                              int n, int co)
{
    const long tid = (long)blockIdx.x * blockDim.x + threadIdx.x;
    if (tid >= (long)n * co) return;
    const int i = (int)(tid / co);
    const int c = (int)(tid - (long)i * co);
    const float cnt = fmaxf(CNT[i], 1.f);
    const float v = ACC[tid] / cnt + Y[(size_t)i * 9 * co + 8 * co + c] + b[c];
    const float h = v > 0.f ? v : (expf(v) - 1.f);   // ELU, alpha=1
    ACC[tid] = h;
    atomicAdd(&stats[c], h);
    atomicAdd(&stats[co + c], h * h);
}

// out = (h - mean) * rsqrt(var + 1e-5) * g + beta   (population stats)
__global__ void bn_apply(const float* __restrict__ H,
                         const float* __restrict__ stats,
                         const float* __restrict__ g,
                         const float* __restrict__ be,
                         float* __restrict__ Hout,
                         int n, int co)
{
    const long tid = (long)blockIdx.x * blockDim.x + threadIdx.x;
    if (tid >= (long)n * co) return;
    const int c = (int)(tid % co);
    const float inv_n = 1.f / (float)n;
    const float m   = stats[c] * inv_n;
    const float var = stats[co + c] * inv_n - m * m;
    Hout[tid] = (H[tid] - m) * rsqrtf(var + 1e-5f) * g[c] + be[c];
}

__global__ void fill_f32(float* __restrict__ p, float v, long n)
{
    const long tid = (long)blockIdx.x * blockDim.x + threadIdx.x;
    if (tid < n) p[tid] = v;
}

__device__ __forceinline__ void atomicMaxF(float* addr, float val)
{
    int* ai  = (int*)addr;
    int  old = __float_as_int(*addr);
    while (__int_as_float(old) < val) {
        const int assumed = old;
        old = atomicCAS(ai, assumed, __float_as_int(val));
        if (old == assumed) break;
    }
}

// segment_max: Hp[cluster[i], c] = max_i H[i, c]
__global__ void pool_max(const float* __restrict__ H,
                         const int*   __restrict__ cluster,
                         float* __restrict__ Hp,
                         int n, int co)
{
    const long tid = (long)blockIdx.x * blockDim.x + threadIdx.x;
    if (tid >= (long)n * co) return;
    const int i = (int)(tid / co);
    const int c = (int)(tid - (long)i * co);
    atomicMaxF(&Hp[(size_t)cluster[i] * co + c], H[tid]);
}

// ---------------------------------------------------------------------------
// Host orchestration
// ---------------------------------------------------------------------------
static inline int cdiv(long a, long b) { return (int)((a + b - 1) / b); }

static void run_layer(const float* Xin, int n, int ci, int co,
                      const float* W, const float* R, const float* b,
                      const float* g, const float* be,
                      const float* ea, const int* ei, int E,
                      float* Hout,
                      float* Ycat, float* ACC, float* CNT, float* stats,
                      hipStream_t stream)
{
    hipMemsetAsync(ACC,   0, (size_t)n * co * sizeof(float), stream);
    hipMemsetAsync(CNT,   0, (size_t)n * sizeof(float),      stream);
    hipMemsetAsync(stats, 0, (size_t)2 * co * sizeof(float), stream);

    const int ncols = 9 * co;
    const long tiles = (long)cdiv(n, 16) * (ncols >> 5);      // 16x32 tiles
    gemm_wmma_f32<<<cdiv(tiles, 4), 128, 0, stream>>>(Xin, W, R, Ycat, n, ci, co);

    const long et = (long)E * (co >> 2);
    edge_accum<<<cdiv(et, 256), 256, 0, stream>>>(Ycat, ea, ei, ACC, CNT, E, co);

    const long nt = (long)n * co;
    node_finalize<<<cdiv(nt, 256), 256, 0, stream>>>(Ycat, ACC, CNT, b, stats, n, co);
    bn_apply<<<cdiv(nt, 256), 256, 0, stream>>>(ACC, stats, g, be, Hout, n, co);
}

extern "C" void kernel_launch(void* const* d_in, const int* in_sizes, int n_in,
                              void* d_out, int out_size, void* d_ws, size_t ws_size,
                              hipStream_t stream)
{
    (void)n_in; (void)ws_size;
    // Input order: x, ea0, ea1, ea2, [W,R,b,g,be]x6, ei0, ei1, ei2, cl1, cl2, n1, n2
    const float* x     = (const float*)d_in[0];
    const float* ea0   = (const float*)d_in[1];
    const float* ea1   = (const float*)d_in[2];
    const float* ea2   = (const float*)d_in[3];
    const int*   ei0   = (const int*)d_in[34];
    const int*   ei1   = (const int*)d_in[35];
    const int*   ei2   = (const int*)d_in[36];
    const int*   cl1   = (const int*)d_in[37];
    const int*   cl2   = (const int*)d_in[38];

    const int N  = in_sizes[0] / 4;
    const int E0 = in_sizes[34] / 2;
    const int E1 = in_sizes[35] / 2;
    const int E2 = in_sizes[36] / 2;
    const int n1 = in_sizes[38];          // cluster2 has n1 entries
    const int n2 = out_size / 256;        // output is [n2, 256]

    const float* P[6][5];                 // W, R, b, g, be per layer
    for (int i = 0; i < 6; ++i)
        for (int j = 0; j < 5; ++j)
            P[i][j] = (const float*)d_in[4 + 5 * i + j];

    auto maxl = [](long a, long b) { return a > b ? a : b; };
    const long szHA   = maxl(maxl((long)N * 32, (long)n1 * 64), (long)n2 * 128);
    const long szHB   = maxl((long)N * 64, maxl((long)n1 * 64, (long)n2 * 64));
    const long szY    = maxl((long)N * 576, (long)n2 * 2304);
    const long szACC  = maxl((long)N * 64, (long)n2 * 256);

    char* p = (char*)d_ws;
    auto carve = [&](long elems) {
        float* r = (float*)p;
        p += (((size_t)elems * sizeof(float)) + 255) & ~(size_t)255;
        return r;
    };
    float* HA    = carve(szHA);
    float* HB    = carve(szHB);
    float* Ycat  = carve(szY);
    float* ACC   = carve(szACC);
    float* CNT   = carve(N);
    float* stats = carve(512);

    // Level 0: layers 1-2 on the full graph
    run_layer(x,  N, 4,  32, P[0][0], P[0][1], P[0][2], P[0][3], P[0][4],
              ea0, ei0, E0, HA, Ycat, ACC, CNT, stats, stream);
    run_layer(HA, N, 32, 64, P[1][0], P[1][1], P[1][2], P[1][3], P[1][4],
              ea0, ei0, E0, HB, Ycat, ACC, CNT, stats, stream);

    // Pool 1: segment_max HB[N,64] -> HA[n1,64]
    fill_f32<<<cdiv((long)n1 * 64, 256), 256, 0, stream>>>(HA, -INFINITY, (long)n1 * 64);
    pool_max<<<cdiv((long)N * 64, 256), 256, 0, stream>>>(HB, cl1, HA, N, 64);

    // Level 1: layers 3-4
    run_layer(HA, n1, 64, 64, P[2][0], P[2][1], P[2][2], P[2][3], P[2][4],
              ea1, ei1, E1, HB, Ycat, ACC, CNT, stats, stream);
    run_layer(HB, n1, 64, 64, P[3][0], P[3][1], P[3][2], P[3][3], P[3][4],
              ea1, ei1, E1, HA, Ycat, ACC, CNT, stats, stream);

    // Pool 2: segment_max HA[n1,64] -> HB[n2,64]
    fill_f32<<<cdiv((long)n2 * 64, 256), 256, 0, stream>>>(HB, -INFINITY, (long)n2 * 64);
    pool_max<<<cdiv((long)n1 * 64, 256), 256, 0, stream>>>(HA, cl2, HB, n1, 64);

    // Level 2: layers 5-6 (layer 6 writes straight to d_out)
    run_layer(HB, n2, 64, 128, P[4][0], P[4][1], P[4][2], P[4][3], P[4][4],
              ea2, ei2, E2, HA, Ycat, ACC, CNT, stats, stream);
    run_layer(HA, n2, 128, 256, P[5][0], P[5][1], P[5][2], P[5][3], P[5][4],
              ea2, ei2, E2, (float*)d_out, Ycat, ACC, CNT, stats, stream);
}